// Contra_balsc_30889404793116
// MI455X (gfx1250) — compile-verified
//
#include <hip/hip_runtime.h>
#include <math.h>

typedef __attribute__((ext_vector_type(2))) float v2f;
typedef __attribute__((ext_vector_type(8))) float v8f;

#define B_ROWS 8192
#define C_CLS  1000
#define DDIM   128
#define NCOL   (B_ROWS + C_CLS)     /* 9192 */
#define NCT    ((NCOL + 15) / 16)   /* 575  */
#define TEMP_INV 10.0f

/* ---------------- Phase A: per-class counts and per-column weights ---------------- */

__global__ void k_init_cnt(int* cnt) {
    int i = blockIdx.x * 256 + threadIdx.x;
    if (i < C_CLS) cnt[i] = 1;            /* each class center contributes 1 */
}

__global__ void k_count_targets(const int* __restrict__ targets, int* __restrict__ cnt) {
    int i = blockIdx.x * 256 + threadIdx.x;
    if (i < B_ROWS) atomicAdd(&cnt[targets[i]], 1);   /* integer atomics: deterministic */
}

__global__ void k_build_cols(const int* __restrict__ targets, const int* __restrict__ cnt,
                             int* __restrict__ colcls, float* __restrict__ wpos,
                             float* __restrict__ wneg) {
    int j = blockIdx.x * 256 + threadIdx.x;
    if (j >= NCOL) return;
    int c = (j < B_ROWS) ? targets[j] : (j - B_ROWS);
    int k = cnt[c];
    colcls[j] = c;
    wneg[j]   = 1.0f / (float)k;
    wpos[j]   = (k > 1) ? (1.0f / (float)(k - 1)) : 0.0f;  /* pos implies k>=2 */
}

/* ---------------- Phase B: fused WMMA GEMM + weighted log-sum-exp ---------------- */

__global__ __launch_bounds__(256) void k_contra_main(
    const float* __restrict__ feat, const float* __restrict__ cent,
    const int* __restrict__ targets, const int* __restrict__ colcls,
    const float* __restrict__ wpos, const float* __restrict__ wneg,
    float* __restrict__ rowout)
{
    const int rowbase = blockIdx.x * 16;
    const int lane  = threadIdx.x & 31;
    const int wave  = threadIdx.x >> 5;
    const int m     = lane & 15;           /* A: M index, B: N index          */
    const int halfk = (lane >> 4) * 2;     /* K sub-offset within each K=4 step */
    const int rhalf = (lane >> 4) * 8;     /* which 8 rows this lane's D holds */

    /* Preload A fragments for all K: 32 x v2f (features rows rowbase..rowbase+15) */
    v2f a[32];
    const float* arow = feat + (size_t)(rowbase + m) * DDIM;
    #pragma unroll
    for (int k = 0; k < 32; ++k)
        a[k] = *(const v2f*)(arow + 4 * k + halfk);

    /* Targets of the 8 rows whose D values this lane owns */
    int rowt[8];
    #pragma unroll
    for (int r = 0; r < 8; ++r)
        rowt[r] = targets[rowbase + rhalf + r];

    float s[8], p[8], n[8];
    #pragma unroll
    for (int r = 0; r < 8; ++r) { s[r] = 0.0f; p[r] = 0.0f; n[r] = 0.0f; }

    for (int ct = wave; ct < NCT; ct += 8) {        /* uniform per wave */
        const int col0 = ct * 16;
        const int j    = col0 + m;                  /* this lane's column */
        const bool valid = (j < NCOL);
        const int jj = valid ? j : (NCOL - 1);      /* clamp loads, mask later */
        const float* brow = (jj < B_ROWS) ? (feat + (size_t)jj * DDIM)
                                          : (cent + (size_t)(jj - B_ROWS) * DDIM);

        v8f acc0 = {}; v8f acc1 = {};               /* two chains to hide WMMA RAW latency */
        #pragma unroll
        for (int k = 0; k < 32; k += 2) {
            v2f b0 = *(const v2f*)(brow + 4 * k + halfk);
            v2f b1 = *(const v2f*)(brow + 4 * (k + 1) + halfk);
            acc0 = __builtin_amdgcn_wmma_f32_16x16x4_f32(
                       false, a[k],     false, b0, (short)0, acc0, false, false);
            acc1 = __builtin_amdgcn_wmma_f32_16x16x4_f32(
                       false, a[k + 1], false, b1, (short)0, acc1, false, false);
        }

        const int   ccls = colcls[jj];
        const float wp   = wpos[jj];
        const float wn   = wneg[jj];

        #pragma unroll
        for (int r = 0; r < 8; ++r) {
            const float raw  = (acc0[r] + acc1[r]) * TEMP_INV;   /* dot / T */
            const int   row  = rowbase + rhalf + r;
            const bool  self = (j == row);
            const bool  pos  = (ccls == rowt[r]) && !self;
            const float e    = __expf(raw) * (pos ? wp : wn);
            s[r] += (valid && !self) ? e    : 0.0f;
            p[r] += (valid && pos)   ? raw  : 0.0f;
            n[r] += (valid && pos)   ? 1.0f : 0.0f;
        }
    }

    /* Reduce across the 16 lanes of each half (columns of the same rows) */
    #pragma unroll
    for (int off = 1; off <= 8; off <<= 1) {
        #pragma unroll
        for (int r = 0; r < 8; ++r) {
            s[r] += __shfl_xor(s[r], off, 32);
            p[r] += __shfl_xor(p[r], off, 32);
            n[r] += __shfl_xor(n[r], off, 32);
        }
    }

    __shared__ float red[8][16][3];
    if ((lane & 15) == 0) {
        #pragma unroll
        for (int r = 0; r < 8; ++r) {
            red[wave][rhalf + r][0] = s[r];
            red[wave][rhalf + r][1] = p[r];
            red[wave][rhalf + r][2] = n[r];
        }
    }
    __syncthreads();

    if (threadIdx.x < 16) {
        float ss = 0.0f, pp = 0.0f, nn = 0.0f;
        #pragma unroll
        for (int w = 0; w < 8; ++w) {
            ss += red[w][threadIdx.x][0];
            pp += red[w][threadIdx.x][1];
            nn += red[w][threadIdx.x][2];
        }
        /* mean log-prob of positives; max-shift cancels exactly, see analysis */
        rowout[rowbase + threadIdx.x] = pp / nn - logf(ss);
    }
}

/* ---------------- Phase C: deterministic final reduction ---------------- */

__global__ __launch_bounds__(256) void k_final_reduce(const float* __restrict__ rowout,
                                                      float* __restrict__ out) {
    __shared__ float sm[256];
    float acc = 0.0f;
    for (int i = threadIdx.x; i < B_ROWS; i += 256) acc += rowout[i];
    sm[threadIdx.x] = acc;
    __syncthreads();
    for (int st = 128; st > 0; st >>= 1) {
        if (threadIdx.x < st) sm[threadIdx.x] += sm[threadIdx.x + st];
        __syncthreads();
    }
    if (threadIdx.x == 0) out[0] = -sm[0] / (float)B_ROWS;
}

/* ---------------- Host launcher ---------------- */

extern "C" void kernel_launch(void* const* d_in, const int* in_sizes, int n_in,
                              void* d_out, int out_size, void* d_ws, size_t ws_size,
                              hipStream_t stream) {
    (void)in_sizes; (void)n_in; (void)out_size; (void)ws_size;

    const float* centers1 = (const float*)d_in[0];   /* [1000, 128] */
    const float* features = (const float*)d_in[1];   /* [8192, 128] */
    const int*   targets  = (const int*)d_in[2];     /* [8192]      */
    float*       out      = (float*)d_out;           /* scalar      */

    /* Workspace layout (256B aligned slices) */
    char* ws = (char*)d_ws;
    int*   cnt    = (int*)(ws + 0);          /* 1024 ints   */
    int*   colcls = (int*)(ws + 4096);       /* 9216 ints   */
    float* wpos   = (float*)(ws + 43008);    /* 9216 floats */
    float* wneg   = (float*)(ws + 81920);    /* 9216 floats */
    float* rowout = (float*)(ws + 120832);   /* 8192 floats */

    k_init_cnt      <<<(C_CLS + 255) / 256, 256, 0, stream>>>(cnt);
    k_count_targets <<<(B_ROWS + 255) / 256, 256, 0, stream>>>(targets, cnt);
    k_build_cols    <<<(NCOL + 255) / 256, 256, 0, stream>>>(targets, cnt, colcls, wpos, wneg);
    k_contra_main   <<<B_ROWS / 16, 256, 0, stream>>>(features, centers1, targets,
                                                      colcls, wpos, wneg, rowout);
    k_final_reduce  <<<1, 256, 0, stream>>>(rowout, out);
}